// AGREE_90872918048956
// MI455X (gfx1250) — compile-verified
//
#include <hip/hip_runtime.h>
#include <hip/hip_bf16.h>

typedef __attribute__((ext_vector_type(2))) float v2f;
typedef __attribute__((ext_vector_type(8))) float v8f;

#define WAVES_PER_BLOCK 4
#define SAMPLES_PER_WAVE 16
#define SAMPLES_PER_BLOCK (WAVES_PER_BLOCK * SAMPLES_PER_WAVE)
#define D 32

// One wave32 processes 16 samples as a 16-row WMMA tile.
//   GEMM1: h1[16,16]  = x[16,128]  @ att_w1[128,16]   (32x V_WMMA_F32_16X16X4_F32)
//   GEMM3: h2[16,16p] = new[16,96] @ pred_w1p[96,16]  (24x V_WMMA_F32_16X16X4_F32)
__global__ __launch_bounds__(128) void agree_fwd_kernel(
    const int* __restrict__ group_inputs,
    const int* __restrict__ item_inputs,
    const int* __restrict__ group_members,
    const float* __restrict__ user_emb,
    const float* __restrict__ item_emb,
    const float* __restrict__ att_w1,   // [128,16]
    const float* __restrict__ att_b1,   // [16]
    const float* __restrict__ att_w2,   // [16,3]
    const float* __restrict__ att_b2,   // [3]
    const float* __restrict__ pred_w1,  // [96,8]
    const float* __restrict__ pred_b1,  // [8]
    const float* __restrict__ pred_w2,  // [8,1]
    const float* __restrict__ pred_b2,  // [1]
    float* __restrict__ out)
{
    __shared__ float w1_s[128 * 16];                          // 8 KB att_w1
    __shared__ float w2p_s[96 * 16];                          // 6 KB pred_w1 padded 8->16 cols
    __shared__ float x_s[WAVES_PER_BLOCK][16][128];           // 32 KB x tile, reused as `new` tile
    __shared__ float h_s[WAVES_PER_BLOCK][16][16];            // 4 KB h1 tile

    const int tid   = threadIdx.x;
    const int wave  = tid >> 5;
    const int lane  = tid & 31;
    const int laneL = lane & 15;
    const int hi    = lane >> 4;          // 0: K+0/K+1 half, 1: K+2/K+3 half

    // ---- preload weight tiles (workgroup cooperative) ----
    for (int i = tid; i < 128 * 16; i += 128) w1_s[i] = att_w1[i];
    for (int i = tid; i < 96 * 16; i += 128) {
        const int r = i >> 4, c = i & 15;
        w2p_s[i] = (c < 8) ? pred_w1[r * 8 + c] : 0.0f;
    }
    __syncthreads();

    const int sbase = blockIdx.x * SAMPLES_PER_BLOCK + wave * SAMPLES_PER_WAVE;

    // ---- per-sample gather indices (lane m holds sample sbase+m; hi half duplicates) ----
    int m0, m1, m2, it;
    {
        const int s   = sbase + laneL;
        const int gid = group_inputs[s];
        it = item_inputs[s];
        const int* gm = group_members + (size_t)gid * 3;
        m0 = gm[0]; m1 = gm[1]; m2 = gm[2];
    }

    // ---- stage x[16][128] = [u0|u1|u2|item] embeds, float4-coalesced ----
    const int seg = lane >> 3;   // 0..3 segment (3 members + item)
    const int o   = lane & 7;    // float4 within 32-float segment
    for (int m = 0; m < 16; ++m) {
        const int u0 = __shfl(m0, m);
        const int u1 = __shfl(m1, m);
        const int u2 = __shfl(m2, m);
        const int ii = __shfl(it, m);
        const float* src;
        if      (seg == 0) src = user_emb + (size_t)u0 * D;
        else if (seg == 1) src = user_emb + (size_t)u1 * D;
        else if (seg == 2) src = user_emb + (size_t)u2 * D;
        else               src = item_emb + (size_t)ii * D;
        const float4 v = ((const float4*)src)[o];
        ((float4*)&x_s[wave][m][0])[lane] = v;   // seg*8+o == lane
    }
    __syncthreads();

    // ---- GEMM1: h1 = x @ att_w1 via 32 chained f32 WMMAs ----
    v8f c1 = {};
    const float* xrow = &x_s[wave][laneL][0];
    #pragma unroll
    for (int kc = 0; kc < 32; ++kc) {
        const int k = kc * 4 + hi * 2;
        v2f a; a.x = xrow[k];               a.y = xrow[k + 1];
        v2f b; b.x = w1_s[k * 16 + laneL];  b.y = w1_s[(k + 1) * 16 + laneL];
        c1 = __builtin_amdgcn_wmma_f32_16x16x4_f32(false, a, false, b,
                                                   (short)0, c1, false, false);
    }

    // bias + relu, park h1 in LDS (C layout: lane holds col laneL, rows r+8*hi)
    const float b1v = att_b1[laneL];
    #pragma unroll
    for (int r = 0; r < 8; ++r) {
        float h = c1[r] + b1v;
        h_s[wave][hi * 8 + r][laneL] = h > 0.0f ? h : 0.0f;
    }
    __syncthreads();

    // ---- attention head: logits[3] = h1_row @ att_w2 + b2, softmax (sample = laneL) ----
    float l0 = att_b2[0], l1 = att_b2[1], l2 = att_b2[2];
    #pragma unroll
    for (int n = 0; n < 16; ++n) {
        const float hv = h_s[wave][laneL][n];
        l0 += hv * att_w2[n * 3 + 0];
        l1 += hv * att_w2[n * 3 + 1];
        l2 += hv * att_w2[n * 3 + 2];
    }
    const float mx = fmaxf(l0, fmaxf(l1, l2));
    const float e0 = __expf(l0 - mx), e1 = __expf(l1 - mx), e2 = __expf(l2 - mx);
    const float inv = 1.0f / (e0 + e1 + e2);
    const float wt0 = e0 * inv, wt1 = e1 * inv, wt2 = e2 * inv;

    // ---- build new = [g*ie | g | ie] in place over x tile (2 lanes per sample) ----
    {
        const int msamp = lane >> 1;
        const int dbase = (lane & 1) * 16;
        const float a0 = __shfl(wt0, msamp);
        const float a1 = __shfl(wt1, msamp);
        const float a2 = __shfl(wt2, msamp);
        float* xr = &x_s[wave][msamp][0];
        #pragma unroll
        for (int d2 = 0; d2 < 16; ++d2) {
            const int d = dbase + d2;
            const float v0 = xr[d], v1 = xr[32 + d], v2 = xr[64 + d], ie = xr[96 + d];
            const float g = a0 * v0 + a1 * v1 + a2 * v2;
            xr[d]      = g * ie;
            xr[32 + d] = g;
            xr[64 + d] = ie;
        }
    }
    __syncthreads();

    // ---- GEMM3: h2 = new[16,96] @ pred_w1p[96,16] via 24 chained f32 WMMAs ----
    v8f c2 = {};
    const float* nrow = &x_s[wave][laneL][0];
    #pragma unroll
    for (int kc = 0; kc < 24; ++kc) {
        const int k = kc * 4 + hi * 2;
        v2f a; a.x = nrow[k];                a.y = nrow[k + 1];
        v2f b; b.x = w2p_s[k * 16 + laneL];  b.y = w2p_s[(k + 1) * 16 + laneL];
        c2 = __builtin_amdgcn_wmma_f32_16x16x4_f32(false, a, false, b,
                                                   (short)0, c2, false, false);
    }

    // ---- final: y = sigmoid(relu(h2 + pb1) @ pred_w2 + pb2), 16-lane shfl reduce ----
    const float pb  = (laneL < 8) ? pred_b1[laneL] : 0.0f;   // pad cols stay exactly 0
    const float pw  = (laneL < 8) ? pred_w2[laneL] : 0.0f;
    const float pb2 = pred_b2[0];
    float ysum[8];
    #pragma unroll
    for (int r = 0; r < 8; ++r) {
        float h2 = c2[r] + pb;
        h2 = h2 > 0.0f ? h2 : 0.0f;
        float p = h2 * pw;
        p += __shfl_xor(p, 1);
        p += __shfl_xor(p, 2);
        p += __shfl_xor(p, 4);
        p += __shfl_xor(p, 8);   // masks < 16: reduction stays inside each 16-lane half
        ysum[r] = p;
    }
    if (laneL == 0) {            // lanes 0 and 16 each emit 8 samples
        #pragma unroll
        for (int r = 0; r < 8; ++r) {
            const float y = 1.0f / (1.0f + __expf(-(ysum[r] + pb2)));
            out[sbase + hi * 8 + r] = y;
        }
    }
}

extern "C" void kernel_launch(void* const* d_in, const int* in_sizes, int n_in,
                              void* d_out, int out_size, void* d_ws, size_t ws_size,
                              hipStream_t stream) {
    const int*   group_inputs  = (const int*)d_in[0];
    const int*   item_inputs   = (const int*)d_in[1];
    const int*   group_members = (const int*)d_in[2];
    const float* user_emb      = (const float*)d_in[3];
    const float* item_emb      = (const float*)d_in[4];
    const float* att_w1        = (const float*)d_in[5];
    const float* att_b1        = (const float*)d_in[6];
    const float* att_w2        = (const float*)d_in[7];
    const float* att_b2        = (const float*)d_in[8];
    const float* pred_w1       = (const float*)d_in[9];
    const float* pred_b1       = (const float*)d_in[10];
    const float* pred_w2       = (const float*)d_in[11];
    const float* pred_b2       = (const float*)d_in[12];
    float* out = (float*)d_out;

    const int B = in_sizes[0];                 // 1048576, divisible by 64
    const int blocks = B / SAMPLES_PER_BLOCK;  // 16384 blocks of 128 threads (4 waves)

    agree_fwd_kernel<<<dim3(blocks), dim3(128), 0, stream>>>(
        group_inputs, item_inputs, group_members, user_emb, item_emb,
        att_w1, att_b1, att_w2, att_b2,
        pred_w1, pred_b1, pred_w2, pred_b2, out);
}